// SpatialGradientLoss3D_32968168964505
// MI455X (gfx1250) — compile-verified
//
#include <hip/hip_runtime.h>

#define D_DIM 64
#define H_DIM 128
#define W_DIM 128
#define TILE  16
#define HALO  18          // TILE + 2 halo
#define NBLOCKS 1024      // 16 maps * 8 * 8 tiles

typedef __attribute__((ext_vector_type(2))) float v2f;
typedef __attribute__((ext_vector_type(8))) float v8f;

// ---------------------------------------------------------------------------
// Main kernel: one block = one 16x16 (H,W) tile of one (b,c) map, marching D.
// Separable Sobel on diff = pred - target, rolling 3-plane ring in LDS.
// ---------------------------------------------------------------------------
__global__ __launch_bounds__(256) void sobel3d_loss_kernel(
    const float* __restrict__ pred, const float* __restrict__ targ,
    float* __restrict__ partial)
{
  __shared__ float bufP[HALO][20];   // raw pred plane (halo), padded stride
  __shared__ float bufT[HALO][20];   // raw target plane (halo)
  __shared__ float sWs[HALO][TILE];  // [1,2,1] along W
  __shared__ float dWs[HALO][TILE];  // [-1,0,1] along W
  __shared__ float u1r[3][256];      // sH(sW)  ring over D
  __shared__ float u2r[3][256];      // dH(sW)
  __shared__ float u3r[3][256];      // sH(dW)
  __shared__ float wsum[8];

  const int tid  = threadIdx.x;
  const int m    = blockIdx.x >> 6;        // map index (b*C+c), 0..15
  const int tile = blockIdx.x & 63;
  const int th0  = (tile >> 3) << 4;
  const int tw0  = (tile & 7) << 4;

  // --- d-invariant load geometry: 324 halo elems, <=2 per thread ---
  const int hh0 = tid / HALO, ww0 = tid % HALO;
  const int gh0 = th0 - 1 + hh0, gw0 = tw0 - 1 + ww0;
  const bool inb0 = ((unsigned)gh0 < (unsigned)H_DIM) && ((unsigned)gw0 < (unsigned)W_DIM);
  const int lin0 = inb0 ? (gh0 * W_DIM + gw0) : 0;
  const unsigned ldsP0 = (unsigned)(unsigned long long)&bufP[hh0][ww0];
  const unsigned ldsT0 = (unsigned)(unsigned long long)&bufT[hh0][ww0];

  const bool has1 = (tid < (HALO * HALO - 256));   // tid < 68
  int lin1 = 0; bool inb1 = false;
  unsigned ldsP1 = 0, ldsT1 = 0;
  int hh1 = 0, ww1 = 0;
  if (has1) {
    const int i1 = tid + 256;
    hh1 = i1 / HALO; ww1 = i1 % HALO;
    const int gh1 = th0 - 1 + hh1, gw1 = tw0 - 1 + ww1;
    inb1 = ((unsigned)gh1 < (unsigned)H_DIM) && ((unsigned)gw1 < (unsigned)W_DIM);
    lin1 = inb1 ? (gh1 * W_DIM + gw1) : 0;
    ldsP1 = (unsigned)(unsigned long long)&bufP[hh1][ww1];
    ldsT1 = (unsigned)(unsigned long long)&bufT[hh1][ww1];
  }

  // Zero-pad entries that async loads never touch (H/W borders): write once.
  if (!inb0) { bufP[hh0][ww0] = 0.0f; bufT[hh0][ww0] = 0.0f; }
  if (has1 && !inb1) { bufP[hh1][ww1] = 0.0f; bufT[hh1][ww1] = 0.0f; }

  // Zero ring (acts as the D = -1 zero plane).
  for (int s = 0; s < 3; ++s) {
    u1r[s][tid] = 0.0f; u2r[s][tid] = 0.0f; u3r[s][tid] = 0.0f;
  }

  float acc = 0.0f;
  const unsigned long long mapBase =
      (unsigned long long)m * (unsigned long long)(D_DIM * H_DIM * W_DIM);

  for (int d = 0; d <= D_DIM; ++d) {
    // --- P1: async-stage raw plane d into LDS (CDNA5 async path) ---
    if (d < D_DIM) {
      const unsigned long long pb =
          mapBase + (unsigned long long)d * (H_DIM * W_DIM);
      if (inb0) {
        unsigned long long aP = (unsigned long long)(pred + pb + lin0);
        unsigned long long aT = (unsigned long long)(targ + pb + lin0);
        asm volatile("global_load_async_to_lds_b32 %0, %1, off"
                     :: "v"(ldsP0), "v"(aP) : "memory");
        asm volatile("global_load_async_to_lds_b32 %0, %1, off"
                     :: "v"(ldsT0), "v"(aT) : "memory");
      }
      if (has1 && inb1) {
        unsigned long long aP = (unsigned long long)(pred + pb + lin1);
        unsigned long long aT = (unsigned long long)(targ + pb + lin1);
        asm volatile("global_load_async_to_lds_b32 %0, %1, off"
                     :: "v"(ldsP1), "v"(aP) : "memory");
        asm volatile("global_load_async_to_lds_b32 %0, %1, off"
                     :: "v"(ldsT1), "v"(aT) : "memory");
      }
      asm volatile("s_wait_asynccnt 0x0" ::: "memory");
    }
    __syncthreads();   // raw ready; previous u-slot reads complete

    const int slot = d % 3;

    // --- P2: W-direction filters on diff (288 items, <=2/thread) ---
    if (d < D_DIM) {
      {
        const int hh = tid >> 4, w = tid & 15;
        const float p0 = bufP[hh][w]     - bufT[hh][w];
        const float p1 = bufP[hh][w + 1] - bufT[hh][w + 1];
        const float p2 = bufP[hh][w + 2] - bufT[hh][w + 2];
        sWs[hh][w] = p0 + 2.0f * p1 + p2;
        dWs[hh][w] = p2 - p0;
      }
      if (tid < (HALO * TILE - 256)) {          // tid < 32
        const int i = tid + 256;
        const int hh = i >> 4, w = i & 15;
        const float p0 = bufP[hh][w]     - bufT[hh][w];
        const float p1 = bufP[hh][w + 1] - bufT[hh][w + 1];
        const float p2 = bufP[hh][w + 2] - bufT[hh][w + 2];
        sWs[hh][w] = p0 + 2.0f * p1 + p2;
        dWs[hh][w] = p2 - p0;
      }
    }
    __syncthreads();   // sW/dW ready

    // --- P3: H-direction filters -> ring slot d%3 (zero plane at d==D) ---
    if (d < D_DIM) {
      const int h = tid >> 4, w = tid & 15;
      const float a0 = sWs[h][w], a1 = sWs[h + 1][w], a2 = sWs[h + 2][w];
      const float b0 = dWs[h][w], b1 = dWs[h + 1][w], b2 = dWs[h + 2][w];
      u1r[slot][tid] = a0 + 2.0f * a1 + a2;   // sH(sW)
      u2r[slot][tid] = a2 - a0;               // dH(sW)
      u3r[slot][tid] = b0 + 2.0f * b1 + b2;   // sH(dW)
    } else {
      u1r[slot][tid] = 0.0f; u2r[slot][tid] = 0.0f; u3r[slot][tid] = 0.0f;
    }
    __syncthreads();   // u[slot] ready

    // --- P4: emit output plane dOut = d-1 (D-direction combine) ---
    if (d >= 1) {
      const int sm1 = (d + 1) % 3;   // (d-2) mod 3 -> plane dOut-1
      const int sc  = (d + 2) % 3;   // (d-1) mod 3 -> plane dOut
      const float gd = u1r[slot][tid] - u1r[sm1][tid];
      const float gh = u2r[sm1][tid] + 2.0f * u2r[sc][tid] + u2r[slot][tid];
      const float gw = u3r[sm1][tid] + 2.0f * u3r[sc][tid] + u3r[slot][tid];
      acc += fabsf(gd) + fabsf(gh) + fabsf(gw);
    }
  }

  // --- block reduction: WMMA fold (lane L + lane L+16) then shfl_xor ---
  // D = ones(16x4) x B + 0, with B[0][n]=acc[lane n], B[2][n]=acc[lane n+16]
  // => every D element of VGPR0 = acc[n] + acc[n+16].
  v2f av; av[0] = 1.0f; av[1] = 1.0f;
  v2f bv; bv[0] = acc;  bv[1] = 0.0f;
  v8f cv = {};
  v8f dv = __builtin_amdgcn_wmma_f32_16x16x4_f32(
      false, av, false, bv, (short)0, cv, false, false);
  float v = dv[0];
  v += __shfl_xor(v, 1, 32);
  v += __shfl_xor(v, 2, 32);
  v += __shfl_xor(v, 4, 32);
  v += __shfl_xor(v, 8, 32);
  if ((tid & 31) == 0) wsum[tid >> 5] = v;
  __syncthreads();
  if (tid == 0) {
    float t = 0.0f;
#pragma unroll
    for (int i = 0; i < 8; ++i) t += wsum[i];
    partial[blockIdx.x] = t;
  }
}

// ---------------------------------------------------------------------------
// Deterministic final reduction of 1024 block partials -> mean.
// ---------------------------------------------------------------------------
__global__ __launch_bounds__(256) void sobel3d_reduce_kernel(
    const float* __restrict__ partial, float* __restrict__ out)
{
  __shared__ float sm[256];
  const int t = threadIdx.x;
  float s = 0.0f;
  for (int i = t; i < NBLOCKS; i += 256) s += partial[i];
  sm[t] = s;
  __syncthreads();
  for (int k = 128; k > 0; k >>= 1) {
    if (t < k) sm[t] += sm[t + k];
    __syncthreads();
  }
  if (t == 0) out[0] = sm[0] * (1.0f / 50331648.0f);  // / (3*4*4*64*128*128)
}

extern "C" void kernel_launch(void* const* d_in, const int* in_sizes, int n_in,
                              void* d_out, int out_size, void* d_ws, size_t ws_size,
                              hipStream_t stream) {
  const float* pred = (const float*)d_in[0];
  const float* targ = (const float*)d_in[1];
  float* out = (float*)d_out;
  float* partial = (float*)d_ws;   // 1024 floats of scratch

  sobel3d_loss_kernel<<<NBLOCKS, 256, 0, stream>>>(pred, targ, partial);
  sobel3d_reduce_kernel<<<1, 256, 0, stream>>>(partial, out);
}